// SpatialConv_18227841204892
// MI455X (gfx1250) — compile-verified
//
#include <hip/hip_runtime.h>

// SCNN-style 4-directional sweep, fp32, as per-step implicit GEMMs on the
// CDNA5 matrix pipe (v_wmma_f32_16x16x4_f32). Previous row/column staged in
// the 320KB WGP LDS with a pair-interleaved layout so each WMMA B fragment is
// a single ds_load_b64; weights pre-transposed into d_ws so each A fragment
// is a single coalesced global_load_b64. LDS staging uses the gfx1250 async
// global->LDS path (ASYNCcnt) when the toolchain exposes it.

typedef float v2f __attribute__((ext_vector_type(2)));
typedef float v8f __attribute__((ext_vector_type(8)));

typedef __attribute__((address_space(1))) int gas_int;   // global (device) ptr
typedef __attribute__((address_space(3))) int las_int;   // LDS (shared) ptr

#define CCH 128                       // channels
#define CH_STRIDE (72 * 200)          // element stride between channels
#define BATCH_STRIDE (128L * 72 * 200)
#define WTE (128 * 128 * 9)           // elements per weight tensor

#if defined(__HIP_DEVICE_COMPILE__) && \
    __has_builtin(__builtin_amdgcn_global_load_async_to_lds_b32) && \
    __has_builtin(__builtin_amdgcn_s_wait_asynccnt)
#define USE_ASYNC 1
#else
#define USE_ASYNC 0
#endif

// One-time repack: w[o][i][k] -> wt[k][i>>1][o][i&1], so a wave's A fragment
// (out-channel o = lane m, K-pair i0..i0+1) is one aligned 8-byte load,
// coalesced across the 16 m-lanes.
__global__ __launch_bounds__(256)
void transpose_weights_kernel(const float* __restrict__ w, float* __restrict__ wt)
{
    int idx = blockIdx.x * 256 + threadIdx.x;
    if (idx < WTE) {
        int k = idx % 9;
        int r = idx / 9;          // r = o*128 + i
        int i = r & 127;
        int o = r >> 7;
        wt[k * 16384 + (i >> 1) * 256 + o * 2 + (i & 1)] = w[idx];
    }
}

// One sweep step: x[cur] += relu(conv1d(x[prev]) + bias) along `npos`
// positions (memory stride pos_stride), K=9, pad=4.
// Grid: 32 blocks (one per batch). Block: 256 threads = 8 waves; wave w owns
// out-channel tile [16w,16w+16). NT = number of 16-wide position tiles.
template<int NT, bool XP>
__global__ __launch_bounds__(256, 1)
void sweep_step_kernel(float* __restrict__ x,
                       const float* __restrict__ wgt,   // XP ? wt layout : [128][128][9]
                       const float* __restrict__ bias,  // [128]
                       int npos, int pos_stride,
                       long prev_off, long cur_off)
{
    extern __shared__ float lds[];   // pair-interleaved: [(c>>1)][col][c&1]
    constexpr int TW = NT * 16;
    constexpr int WL = TW + 8;       // columns incl. 4+4 halo

    const long base = (long)blockIdx.x * BATCH_STRIDE;
    const int  tid  = threadIdx.x;

    // ---- stage prev row/column into LDS ----
#if USE_ASYNC
    for (int idx = tid; idx < CCH * TW; idx += 256) {
        int c = idx / TW;
        int p = idx - c * TW;
        float* dst = &lds[((c >> 1) * WL + 4 + p) * 2 + (c & 1)];
        if (p < npos) {
            const float* src =
                &x[base + prev_off + (long)c * CH_STRIDE + (long)p * pos_stride];
            __builtin_amdgcn_global_load_async_to_lds_b32(
                (gas_int*)src, (las_int*)dst, 0, 0);
        } else {
            *dst = 0.0f;
        }
    }
#else
    for (int idx = tid; idx < (CCH / 2) * TW; idx += 256) {
        int pr = idx / TW;
        int p  = idx - pr * TW;
        v2f v = (v2f){0.0f, 0.0f};
        if (p < npos) {
            long g = base + prev_off + (long)(pr * 2) * CH_STRIDE
                   + (long)p * pos_stride;
            v.x = x[g];
            v.y = x[g + CH_STRIDE];
        }
        *(v2f*)&lds[(pr * WL + 4 + p) * 2] = v;   // ds_store_b64
    }
#endif
    // zero the halo columns
    for (int idx = tid; idx < (CCH / 2) * 8; idx += 256) {
        int pr = idx >> 3, j = idx & 7;
        int col = (j < 4) ? j : TW + j;
        *(v2f*)&lds[(pr * WL + col) * 2] = (v2f){0.0f, 0.0f};
    }
#if USE_ASYNC
    __builtin_amdgcn_s_wait_asynccnt(0);
#endif
    __syncthreads();

    const int lane = tid & 31;
    const int mo   = (tid >> 5) * 16;   // wave's out-channel tile base
    const int mn   = lane & 15;         // A row / B,D column for this lane
    const int hl   = lane >> 4;         // lane half: K sub-pair / M half

    v8f acc[NT];
#pragma unroll
    for (int t = 0; t < NT; ++t) acc[t] = (v8f){0, 0, 0, 0, 0, 0, 0, 0};

    // reduction over (tap k, in-channel i), K=4 per WMMA
#pragma unroll 1
    for (int k = 0; k < 9; ++k) {
        const float* aK = wgt + (size_t)k * 16384 + hl * 256 + (mo + mn) * 2;
        const float* bK = lds + (mn + k) * 2;
#pragma unroll 1
        for (int ci = 0; ci < CCH / 4; ++ci) {
            v2f a;
            if constexpr (XP) {
                a = *(const v2f*)(aK + ci * 512);           // global_load_b64
            } else {
                const float* wr = wgt + (size_t)(mo + mn) * (CCH * 9);
                int i0 = ci * 4 + hl * 2;
                a.x = wr[(size_t)i0 * 9 + k];
                a.y = wr[(size_t)(i0 + 1) * 9 + k];
            }
            const float* bp = bK + (ci * 2 + hl) * (WL * 2);
#pragma unroll
            for (int t = 0; t < NT; ++t) {
                v2f bb = *(const v2f*)(bp + t * 32);        // ds_load_b64
                acc[t] = __builtin_amdgcn_wmma_f32_16x16x4_f32(
                    false, a, false, bb, (short)0, acc[t], false, false);
            }
        }
    }

    // ---- epilogue: out = cur + relu(acc + bias) ----
    float bv[8];
#pragma unroll
    for (int j = 0; j < 8; ++j) bv[j] = bias[mo + hl * 8 + j];

#pragma unroll
    for (int t = 0; t < NT; ++t) {
        int p = t * 16 + mn;
        if (p < npos) {
            long g0 = base + cur_off + (long)p * pos_stride
                    + (long)(mo + hl * 8) * CH_STRIDE;
#pragma unroll
            for (int j = 0; j < 8; ++j) {   // D layout: VGPR j -> M = j + 8*hl
                long ga = g0 + (long)j * CH_STRIDE;
                float cur = x[ga];
                float v = acc[t][j] + bv[j];
                v = v > 0.0f ? v : 0.0f;
                x[ga] = cur + v;
            }
        }
    }
}

extern "C" void kernel_launch(void* const* d_in, const int* in_sizes, int n_in,
                              void* d_out, int out_size, void* d_ws, size_t ws_size,
                              hipStream_t stream)
{
    const float* x = (const float*)d_in[0];
    const float* W[4]  = {(const float*)d_in[1], (const float*)d_in[3],
                          (const float*)d_in[5], (const float*)d_in[7]};
    const float* Bv[4] = {(const float*)d_in[2], (const float*)d_in[4],
                          (const float*)d_in[6], (const float*)d_in[8]};
    float* out = (float*)d_out;

    const int B = 32, H = 72, Wd = 200;
    size_t nelem = (size_t)B * CCH * H * Wd;
    (void)hipMemcpyAsync(out, x, nelem * sizeof(float),
                         hipMemcpyDeviceToDevice, stream);

    // pre-transpose weights into workspace (once per launch) if it fits
    const bool XP = ws_size >= (size_t)4 * WTE * sizeof(float);
    const float* WT[4] = {W[0], W[1], W[2], W[3]};
    if (XP) {
        float* wsf = (float*)d_ws;
        for (int i = 0; i < 4; ++i) {
            float* wt = wsf + (size_t)i * WTE;
            transpose_weights_kernel<<<(WTE + 255) / 256, 256, 0, stream>>>(W[i], wt);
            WT[i] = wt;
        }
    }

    dim3 grid(B), block(256);
    constexpr int NT_W = 13;          // ceil(200/16) tiles (H sweeps)
    constexpr int NT_H = 5;           // ceil(72/16)  tiles (W sweeps)
    size_t ldsW = (size_t)CCH * (NT_W * 16 + 8) * sizeof(float);  // ~110.6 KB
    size_t ldsH = (size_t)CCH * (NT_H * 16 + 8) * sizeof(float);  // ~45 KB

    auto stepW = [&](int d, long po, long co) {   // H sweeps: conv along W
        if (XP) sweep_step_kernel<NT_W, true ><<<grid, block, ldsW, stream>>>(
                    out, WT[d], Bv[d], Wd, 1, po, co);
        else    sweep_step_kernel<NT_W, false><<<grid, block, ldsW, stream>>>(
                    out, WT[d], Bv[d], Wd, 1, po, co);
    };
    auto stepH = [&](int d, long po, long co) {   // W sweeps: conv along H
        if (XP) sweep_step_kernel<NT_H, true ><<<grid, block, ldsH, stream>>>(
                    out, WT[d], Bv[d], H, Wd, po, co);
        else    sweep_step_kernel<NT_H, false><<<grid, block, ldsH, stream>>>(
                    out, WT[d], Bv[d], H, Wd, po, co);
    };

    // down sweep (axis H, forward)
    for (int h = 1; h < H; ++h)      stepW(0, (long)(h - 1) * Wd, (long)h * Wd);
    // up sweep (axis H, reverse)
    for (int h = H - 2; h >= 0; --h) stepW(1, (long)(h + 1) * Wd, (long)h * Wd);
    // right sweep (axis W, forward)
    for (int w = 1; w < Wd; ++w)     stepH(2, (long)(w - 1), (long)w);
    // left sweep (axis W, reverse)
    for (int w = Wd - 2; w >= 0; --w) stepH(3, (long)(w + 1), (long)w);
}